// Whitening2d_7327214207107
// MI455X (gfx1250) — compile-verified
//
#include <hip/hip_runtime.h>

typedef __attribute__((ext_vector_type(2))) float v2f;
typedef __attribute__((ext_vector_type(8))) float v8f;

#define BATCH 32
#define SEQ   2048
#define FEAT  512
#define MOM   0.1f
#define EPSV  1e-5f
#define TILE_K 16
#define KP    18   // padded K row stride (floats): even (b64-aligned) + conflict-free

enum { M_COV = 0, M_MUL = 1, M_NSUPD = 2, M_DECOR = 3 };

// ---------------------------------------------------------------------------
// mean[b,f] = 0.9*running_mean[f] + 0.1*mean_s x[b,s,f]
// ---------------------------------------------------------------------------
__global__ __launch_bounds__(256) void mean_kernel(const float* __restrict__ x,
                                                   const float* __restrict__ rmean,
                                                   float* __restrict__ mean) {
    int b = blockIdx.y;
    int f = blockIdx.x * 256 + threadIdx.x;
    const float* xp = x + (size_t)b * SEQ * FEAT + f;
    float s = 0.0f;
    for (int ss = 0; ss < SEQ; ++ss) s += xp[(size_t)ss * FEAT];
    mean[b * FEAT + f] = (1.0f - MOM) * rmean[f] + MOM * (s * (1.0f / (float)SEQ));
}

// ---------------------------------------------------------------------------
// tr[b] = trace(sigma[b])
// ---------------------------------------------------------------------------
__global__ __launch_bounds__(256) void trace_kernel(const float* __restrict__ sigma,
                                                    float* __restrict__ tr) {
    int b = blockIdx.x;
    int t = threadIdx.x;
    float s = 0.0f;
    for (int f = t; f < FEAT; f += 256) s += sigma[((size_t)b * FEAT + f) * FEAT + f];
    __shared__ float red[256];
    red[t] = s;
    __syncthreads();
    for (int off = 128; off > 0; off >>= 1) {
        if (t < off) red[t] += red[t + off];
        __syncthreads();
    }
    if (t == 0) tr[b] = red[0];
}

// ---------------------------------------------------------------------------
// P[b] = I
// ---------------------------------------------------------------------------
__global__ __launch_bounds__(256) void eye_kernel(float* __restrict__ P) {
    size_t idx = (size_t)blockIdx.x * 256 + threadIdx.x;
    int ij = (int)(idx & ((size_t)FEAT * FEAT - 1));
    P[idx] = ((ij >> 9) == (ij & (FEAT - 1))) ? 1.0f : 0.0f;
}

// ---------------------------------------------------------------------------
// Batched tiled GEMM on v_wmma_f32_16x16x4_f32.
// Workgroup: 256 threads (8 wave32) -> 128x128 C tile, waves 2(M)x4(N),
// each wave: 4x2 subtiles of 16x16 (8 v8f accumulators).
// LDS layout is K-innermost (lA[m][KP], lB[n][KP]) so each lane's K-pair
// fragment is a single aligned ds_load_b64 into an adjacent VGPR pair.
//   M_COV  : C = blend(rcov, (x-m)^T (x-m) / (S-1) + eps I)    K = SEQ
//   M_MUL  : D = A @ B                                          K = FEAT
//   M_NSUPD: P = 1.5*P - (0.5/tr[b]) * (T2 @ sigma)  (in place) K = FEAT
//   M_DECOR: out = ((x-m) @ P) * rsqrt(tr[b])                   K = FEAT
// ---------------------------------------------------------------------------
template <int MODE>
__global__ __launch_bounds__(256) void gemm_kernel(const float* __restrict__ Ap,
                                                   const float* __restrict__ Bp,
                                                   const float* __restrict__ Ep,
                                                   float* __restrict__ Dp,
                                                   const float* __restrict__ meanp,
                                                   const float* __restrict__ trp) {
    constexpr int KDIM = (MODE == M_COV) ? SEQ : FEAT;

    const int b    = blockIdx.z;
    const int tid  = threadIdx.x;
    const int wave = tid >> 5;
    const int lane = tid & 31;
    const int wm   = (wave & 1) * 64;   // wave offset in M within 128
    const int wn   = (wave >> 1) * 32;  // wave offset in N within 128
    const int hl   = lane >> 4;         // half-wave select (K pairs / M high)
    const int ln   = lane & 15;

    const int n0 = blockIdx.x * 128;
    const int m0 = blockIdx.y * 128;

    __shared__ float lA[128][KP];
    __shared__ float lB[128][KP];

    const v8f vzero = {0.f, 0.f, 0.f, 0.f, 0.f, 0.f, 0.f, 0.f};
    v8f acc[4][2];
#pragma unroll
    for (int i = 0; i < 4; ++i)
#pragma unroll
        for (int j = 0; j < 2; ++j) acc[i][j] = vzero;

    for (int kt = 0; kt < KDIM; kt += TILE_K) {
        // ---- stage A tile (lA[m][k]) and B tile (lB[n][k]) ----
        if constexpr (MODE == M_COV) {
            // A[m,k] = xn[b, kt+k, m0+m]  (f contiguous in global -> coalesced)
#pragma unroll
            for (int i = 0; i < 8; ++i) {
                int e = tid + i * 256, k = e >> 7, m = e & 127;
                int f = m0 + m;
                lA[m][k] = Ap[((size_t)b * SEQ + kt + k) * FEAT + f] - meanp[b * FEAT + f];
            }
#pragma unroll
            for (int i = 0; i < 8; ++i) {
                int e = tid + i * 256, k = e >> 7, n = e & 127;
                int g = n0 + n;
                lB[n][k] = Bp[((size_t)b * SEQ + kt + k) * FEAT + g] - meanp[b * FEAT + g];
            }
        } else if constexpr (MODE == M_DECOR) {
            // A[m,k] = xn[b, m0+m, kt+k]
#pragma unroll
            for (int i = 0; i < 8; ++i) {
                int e = tid + i * 256, m = e >> 4, k = e & 15;
                lA[m][k] = Ap[((size_t)b * SEQ + m0 + m) * FEAT + kt + k]
                         - meanp[b * FEAT + kt + k];
            }
#pragma unroll
            for (int i = 0; i < 8; ++i) {
                int e = tid + i * 256, k = e >> 7, n = e & 127;
                lB[n][k] = Bp[((size_t)b * FEAT + kt + k) * FEAT + n0 + n];
            }
        } else {  // M_MUL / M_NSUPD : both operands are [B, FEAT, FEAT] row-major
#pragma unroll
            for (int i = 0; i < 8; ++i) {
                int e = tid + i * 256, m = e >> 4, k = e & 15;
                lA[m][k] = Ap[((size_t)b * FEAT + m0 + m) * FEAT + kt + k];
            }
#pragma unroll
            for (int i = 0; i < 8; ++i) {
                int e = tid + i * 256, k = e >> 7, n = e & 127;
                lB[n][k] = Bp[((size_t)b * FEAT + kt + k) * FEAT + n0 + n];
            }
        }
        __syncthreads();

        // ---- 16-deep K slab = 4 wmma k-chunks; 32 WMMA per wave per slab ----
#pragma unroll
        for (int kc = 0; kc < 4; ++kc) {
            const int kk = kc * 4 + 2 * hl;  // even -> 8B-aligned b64 LDS reads
            v2f bf0 = *(const v2f*)&lB[wn + 0 * 16 + ln][kk];
            v2f bf1 = *(const v2f*)&lB[wn + 1 * 16 + ln][kk];
#pragma unroll
            for (int i = 0; i < 4; ++i) {
                v2f af = *(const v2f*)&lA[wm + i * 16 + ln][kk];
                acc[i][0] = __builtin_amdgcn_wmma_f32_16x16x4_f32(
                    false, af, false, bf0, (short)0, acc[i][0], false, false);
                acc[i][1] = __builtin_amdgcn_wmma_f32_16x16x4_f32(
                    false, af, false, bf1, (short)0, acc[i][1], false, false);
            }
        }
        __syncthreads();
    }

    // ---- epilogue ----
    float scale = 0.0f;
    if constexpr (MODE == M_NSUPD) scale = -0.5f / trp[b];
    if constexpr (MODE == M_DECOR) scale = rsqrtf(trp[b]);

#pragma unroll
    for (int i = 0; i < 4; ++i)
#pragma unroll
        for (int j = 0; j < 2; ++j)
#pragma unroll
            for (int v = 0; v < 8; ++v) {
                int gM = m0 + wm + i * 16 + v + 8 * hl;  // C layout: M = vgpr + 8*(lane/16)
                int gN = n0 + wn + j * 16 + ln;          //           N = lane%16
                float val = acc[i][j][v];
                if constexpr (MODE == M_COV) {
                    float sr = val * (1.0f / (float)(SEQ - 1));
                    if (gM == gN) sr += EPSV;
                    Dp[((size_t)b * FEAT + gM) * FEAT + gN] =
                        (1.0f - MOM) * Ep[gM * FEAT + gN] + MOM * sr;
                } else if constexpr (MODE == M_MUL) {
                    Dp[((size_t)b * FEAT + gM) * FEAT + gN] = val;
                } else if constexpr (MODE == M_NSUPD) {
                    size_t idx = ((size_t)b * FEAT + gM) * FEAT + gN;
                    Dp[idx] = 1.5f * Ep[idx] + scale * val;   // in-place P update
                } else {  // M_DECOR
                    Dp[((size_t)b * SEQ + gM) * FEAT + gN] = val * scale;
                }
            }
}

// ---------------------------------------------------------------------------
extern "C" void kernel_launch(void* const* d_in, const int* in_sizes, int n_in,
                              void* d_out, int out_size, void* d_ws, size_t ws_size,
                              hipStream_t stream) {
    (void)in_sizes; (void)n_in; (void)out_size; (void)ws_size;

    const float* x     = (const float*)d_in[0];  // [32,2048,512]
    const float* rmean = (const float*)d_in[1];  // [512]
    const float* rcov  = (const float*)d_in[2];  // [512,512]
    float*       out   = (float*)d_out;          // [32,2048,512]

    char* ws = (char*)d_ws;
    const size_t MATB = (size_t)BATCH * FEAT * FEAT * sizeof(float);  // 32 MB
    float* mean  = (float*)ws;                          // 32*512 f32 = 64 KB
    float* tr    = (float*)(ws + 65536);                // 32 f32
    float* sigma = (float*)(ws + 131072);
    float* P     = (float*)(ws + 131072 + 1 * MATB);
    float* T1    = (float*)(ws + 131072 + 2 * MATB);
    float* T2    = (float*)(ws + 131072 + 3 * MATB);

    dim3 blk(256);

    mean_kernel<<<dim3(FEAT / 256, BATCH), blk, 0, stream>>>(x, rmean, mean);

    gemm_kernel<M_COV><<<dim3(FEAT / 128, FEAT / 128, BATCH), blk, 0, stream>>>(
        x, x, rcov, sigma, mean, nullptr);

    trace_kernel<<<dim3(BATCH), blk, 0, stream>>>(sigma, tr);

    eye_kernel<<<dim3((BATCH * FEAT * FEAT) / 256), blk, 0, stream>>>(P);

    for (int it = 0; it < 4; ++it) {
        gemm_kernel<M_MUL><<<dim3(FEAT / 128, FEAT / 128, BATCH), blk, 0, stream>>>(
            P, P, nullptr, T1, nullptr, nullptr);
        gemm_kernel<M_MUL><<<dim3(FEAT / 128, FEAT / 128, BATCH), blk, 0, stream>>>(
            T1, P, nullptr, T2, nullptr, nullptr);
        gemm_kernel<M_NSUPD><<<dim3(FEAT / 128, FEAT / 128, BATCH), blk, 0, stream>>>(
            T2, sigma, P, P, nullptr, tr);
    }

    gemm_kernel<M_DECOR><<<dim3(FEAT / 128, SEQ / 128, BATCH), blk, 0, stream>>>(
        x, P, nullptr, out, mean, tr);
}